// AttentionUnit_9019431321918
// MI455X (gfx1250) — compile-verified
//
#include <hip/hip_runtime.h>

// AttentionUnit, algebraically collapsed:
//   s[b,e,v] = K * e[e] * v[v],  K = dot(w_g, w_f)   (rank-1 logits!)
//   t(e,v) = exp(K e v);  Z[v] = col-sums; W[e] = row-sums
//   out0 = v + M * (t    @ (v/Z)),  M = dot(w_h, w_m)
//   out1 = e + N * (t^T  @ (e/W)),  N = dot(w_l, w_n)
// Three D x D sweeps per batch; exp2 on the TRANS pipe (log2e folded into the
// hoisted row coefficient), all reductions on the matrix pipe via
// V_WMMA_F32_16X16X4_F32 (B col0 = ones -> sums, col1 = weights -> weighted
// sums, both in one accumulator). Each wave processes its two 16-row tiles in
// one column sweep so every LDS load / B-build feeds two WMMAs.

typedef __attribute__((ext_vector_type(2))) float v2f;
typedef __attribute__((ext_vector_type(8))) float v8f;

#define NB  512   // batch
#define DIM 512   // e / v length

#if __has_builtin(__builtin_amdgcn_exp2f)
#define EXP2(x) __builtin_amdgcn_exp2f(x)
#else
#define EXP2(x) __expf((x) * 0.69314718055994531f)
#endif

__device__ __forceinline__ v8f wmma4(v2f a, v2f b, v8f c) {
  return __builtin_amdgcn_wmma_f32_16x16x4_f32(false, a, false, b,
                                               (short)0, c, false, false);
}

// For each row r in xs: s0[r] = sum_c t, s1[r] = sum_c ps[c]*t,
// with t = exp2(kl2e * xs[r] * ys[c]) over all DIM columns.
// Each wave owns rows [wave*16, wave*16+16) and [(wave+16)*16, ...+16).
__device__ __forceinline__ void sweep(const float* __restrict__ xs,
                                      const float* __restrict__ ys,
                                      const float* __restrict__ ps,
                                      float* __restrict__ s0,
                                      float* __restrict__ s1, float kl2e) {
  const int lane = threadIdx.x & 31;          // wave32
  const int wave = threadIdx.x >> 5;          // 16 waves
  const int m    = lane & 15;                 // A row / B-D column index
  const int half = lane >> 4;
  const int K0   = half << 1;                 // K-pair {0,1} or {2,3}

  // Lane-constant B-build coefficients: b = wm * p + base
  const float base = (m == 0) ? 1.0f : 0.0f;  // column 0: plain sum
  const float wm   = (m == 1) ? 1.0f : 0.0f;  // column 1: weighted sum

  const int rb0 = wave * 16;                  // tile A rows
  const int rb1 = (wave + 16) * 16;           // tile B rows
  const float kx0 = kl2e * xs[rb0 + m];
  const float kx1 = kl2e * xs[rb1 + m];

  v8f a00 = {}, a01 = {}, a10 = {}, a11 = {};  // 4 chains hide WMMA latency
  for (int vc = 0; vc < DIM; vc += 8) {
    // ---- column chunk vc .. vc+3 (shared loads + B for both row tiles) ----
    float2 yv = *(const float2*)(ys + vc + K0);       // ds_load_b64
    float2 pv = *(const float2*)(ps + vc + K0);
    v2f b0; b0.x = fmaf(wm, pv.x, base); b0.y = fmaf(wm, pv.y, base);
    v2f aA; aA.x = EXP2(kx0 * yv.x); aA.y = EXP2(kx0 * yv.y);
    v2f aB; aB.x = EXP2(kx1 * yv.x); aB.y = EXP2(kx1 * yv.y);
    a00 = wmma4(aA, b0, a00);
    a10 = wmma4(aB, b0, a10);
    // ---- column chunk vc+4 .. vc+7 ----
    float2 yw = *(const float2*)(ys + vc + 4 + K0);
    float2 pw = *(const float2*)(ps + vc + 4 + K0);
    v2f b1; b1.x = fmaf(wm, pw.x, base); b1.y = fmaf(wm, pw.y, base);
    v2f aC; aC.x = EXP2(kx0 * yw.x); aC.y = EXP2(kx0 * yw.y);
    v2f aD; aD.x = EXP2(kx1 * yw.x); aD.y = EXP2(kx1 * yw.y);
    a01 = wmma4(aC, b1, a01);
    a11 = wmma4(aD, b1, a11);
  }
  // C/D layout: lane l, VGPR j -> D[M = j + (l<16?0:8), N = l%16]
  if (m < 2) {
    float* dst = (m == 0) ? s0 : s1;
    const int off = half ? 8 : 0;
#pragma unroll
    for (int j = 0; j < 8; ++j) {
      dst[rb0 + off + j] = a00[j] + a01[j];
      dst[rb1 + off + j] = a10[j] + a11[j];
    }
  }
}

__global__ __launch_bounds__(DIM)
void attention_unit_kernel(const float* __restrict__ v_in,
                           const float* __restrict__ e_in,
                           const float* __restrict__ w_f,
                           const float* __restrict__ w_g,
                           const float* __restrict__ w_h,
                           const float* __restrict__ w_l,
                           const float* __restrict__ w_m,
                           const float* __restrict__ w_n,
                           float* __restrict__ out_v,
                           float* __restrict__ out_e) {
  __shared__ __align__(16) float e_sh[DIM];
  __shared__ __align__(16) float v_sh[DIM];
  __shared__ __align__(16) float s0_sh[DIM];  // Z / W (row sums)
  __shared__ __align__(16) float s1_sh[DIM];  // weighted sums
  __shared__ __align__(16) float p_sh[DIM];   // per-column weights

  const int b   = blockIdx.x;
  const int tid = threadIdx.x;

  v_sh[tid] = v_in[b * DIM + tid];
  e_sh[tid] = e_in[b * DIM + tid];

  // Scalar collapses of the four channel lifts (redundant per thread, cached).
  float K = 0.f, Mw = 0.f, Nw = 0.f;
#pragma unroll
  for (int i = 0; i < 16; ++i) {
    K  += w_g[i] * w_f[i];
    Mw += w_h[i] * w_m[i];
    Nw += w_l[i] * w_n[i];
  }
  const float kl2e = K * 1.4426950408889634f;   // fold log2(e) into coefficient
  __syncthreads();

  // Sweep 1: rows = v, cols = e  ->  s0 = Z[v] (softmax-over-e denominators)
  sweep(v_sh, e_sh, e_sh, s0_sh, s1_sh, kl2e);
  __syncthreads();
  p_sh[tid] = v_sh[tid] / s0_sh[tid];           // u[v] = v/Z
  __syncthreads();

  // Sweep 2: rows = e, cols = v, weights u  ->  s0 = W[e], s1 = o_v[e]/M
  sweep(e_sh, v_sh, p_sh, s0_sh, s1_sh, kl2e);
  __syncthreads();
  out_v[b * DIM + tid] = v_sh[tid] + Mw * s1_sh[tid];
  const float we = e_sh[tid] / s0_sh[tid];      // e/W
  __syncthreads();
  p_sh[tid] = we;
  __syncthreads();

  // Sweep 3: rows = v, cols = e, weights e/W  ->  s1 = o_e[v]/N
  sweep(v_sh, e_sh, p_sh, s0_sh, s1_sh, kl2e);
  __syncthreads();
  out_e[b * DIM + tid] = e_sh[tid] + Nw * s1_sh[tid];
}

extern "C" void kernel_launch(void* const* d_in, const int* in_sizes, int n_in,
                              void* d_out, int out_size, void* d_ws, size_t ws_size,
                              hipStream_t stream) {
  const float* v_in = (const float*)d_in[0];
  const float* e_in = (const float*)d_in[1];
  const float* w_f  = (const float*)d_in[2];
  const float* w_g  = (const float*)d_in[3];
  const float* w_h  = (const float*)d_in[4];
  const float* w_l  = (const float*)d_in[5];
  const float* w_m  = (const float*)d_in[6];
  const float* w_n  = (const float*)d_in[7];
  float* out = (float*)d_out;

  attention_unit_kernel<<<dim3(NB), dim3(DIM), 0, stream>>>(
      v_in, e_in, w_f, w_g, w_h, w_l, w_m, w_n,
      out, out + (size_t)NB * DIM);
}